// BatchRGATLayer_30777735643852
// MI455X (gfx1250) — compile-verified
//
#include <hip/hip_runtime.h>
#include <hip/hip_bf16.h>

typedef __attribute__((ext_vector_type(2))) float v2f;
typedef __attribute__((ext_vector_type(4))) float v4f;
typedef __attribute__((ext_vector_type(8))) float v8f;

#define ALPHA_LRELU 0.2f
#define NEG_INF_F  (-1.0e30f)

// ---------------------------------------------------------------------------
// Kernel 0: w1a3[e'] = sum_c W1[e'][c] * a3[c]      (W1: 64x256, a3 = a[512:768])
// ---------------------------------------------------------------------------
__global__ void w1a3_kernel(const float* __restrict__ W1, const float* __restrict__ a,
                            float* __restrict__ w1a3) {
    int t = threadIdx.x;            // 64 threads
    const float* row = W1 + (size_t)t * 256;
    float acc = 0.f;
    #pragma unroll 4
    for (int c = 0; c < 256; c += 4) {
        v4f v = *(const v4f*)(row + c);
        acc += v[0] * a[512 + c] + v[1] * a[512 + c + 1] +
               v[2] * a[512 + c + 2] + v[3] * a[512 + c + 3];
    }
    w1a3[t] = acc;
}

// ---------------------------------------------------------------------------
// Kernel 1: h = x @ W  via V_WMMA_F32_16X16X4_F32.
// grid (32, 4); block 256 = 8 waves. Block computes a 16x256 slab of h.
// Wave w computes N-tiles w and w+8 (16 cols each), K = 256 -> 64 WMMA steps.
// ---------------------------------------------------------------------------
__global__ void gemm_h_kernel(const float* __restrict__ x, const float* __restrict__ W,
                              float* __restrict__ h) {
    int b    = blockIdx.y;
    int i0   = blockIdx.x * 16;
    int wave = threadIdx.x >> 5;
    int lane = threadIdx.x & 31;

    // A fragment layout (16x4 f32): lanes 0-15 -> K={k0,k0+1}, lanes 16-31 -> K={k0+2,k0+3}
    int M    = lane & 15;
    int koff = (lane >> 4) << 1;
    // B fragment layout (4x16 f32): vgpr0 K = k0 + (lane<16?0:2), vgpr1 = +1, N = lane%16
    int klo  = (lane < 16) ? 0 : 2;
    int ncol = lane & 15;

    const float* xrow = x + (size_t)(b * 512 + i0 + M) * 256;
    int n0a = wave * 16;
    int n0b = (wave + 8) * 16;

    v8f c0 = {}; v8f c1 = {};
    for (int k0 = 0; k0 < 256; k0 += 4) {
        v2f afrag = *(const v2f*)(xrow + k0 + koff);   // contiguous K pair

        const float* wp = W + (size_t)(k0 + klo) * 256;
        v2f b0, b1;
        b0[0] = wp[n0a + ncol]; b0[1] = wp[256 + n0a + ncol];
        b1[0] = wp[n0b + ncol]; b1[1] = wp[256 + n0b + ncol];

        c0 = __builtin_amdgcn_wmma_f32_16x16x4_f32(false, afrag, false, b0, (short)0, c0, false, false);
        c1 = __builtin_amdgcn_wmma_f32_16x16x4_f32(false, afrag, false, b1, (short)0, c1, false, false);
    }

    // C layout: vgpr v -> row v + (lane>=16 ? 8 : 0), col = lane%16
    int mbase = (lane >> 4) << 3;
    float* hb = h + (size_t)(b * 512 + i0) * 256;
    #pragma unroll
    for (int v = 0; v < 8; ++v) {
        hb[(size_t)(mbase + v) * 256 + n0a + ncol] = c0[v];
        hb[(size_t)(mbase + v) * 256 + n0b + ncol] = c1[v];
    }
}

// ---------------------------------------------------------------------------
// Kernel 2: s_i[b,n] = h[b,n,:].a1 ; s_j[b,n] = h[b,n,:].a2
// ---------------------------------------------------------------------------
__global__ void sij_kernel(const float* __restrict__ h, const float* __restrict__ a,
                           float* __restrict__ s_i, float* __restrict__ s_j) {
    int idx = blockIdx.x * blockDim.x + threadIdx.x;  // 0..2047
    const float* hr = h + (size_t)idx * 256;
    float acc1 = 0.f, acc2 = 0.f;
    #pragma unroll 4
    for (int o = 0; o < 256; o += 4) {
        v4f hv = *(const v4f*)(hr + o);
        acc1 += hv[0] * a[o]       + hv[1] * a[o + 1]       + hv[2] * a[o + 2]       + hv[3] * a[o + 3];
        acc2 += hv[0] * a[256 + o] + hv[1] * a[256 + o + 1] + hv[2] * a[256 + o + 2] + hv[3] * a[256 + o + 3];
    }
    s_i[idx] = acc1;
    s_j[idx] = acc2;
}

// ---------------------------------------------------------------------------
// Kernel 3 (HBM-bound stream, max occupancy): one block per (b, i) row.
// 2048 blocks x 8 waves stream edge (268 MB) with non-temporal b128 loads,
// compute masked leaky-relu scores e[b,i,j] -> workspace (4 MB, L2-resident).
// ---------------------------------------------------------------------------
__global__ void score_kernel(const float* __restrict__ edge, const float* __restrict__ adj,
                             const float* __restrict__ s_i, const float* __restrict__ s_j,
                             const float* __restrict__ w1a3, float* __restrict__ escore) {
    __shared__ float wle[64];
    int b   = blockIdx.y;
    int i   = blockIdx.x;
    int tid = threadIdx.x;

    if (tid < 64) wle[tid] = w1a3[tid];
    __syncthreads();

    float si = s_i[b * 512 + i];          // uniform -> scalar load
    const float* ebase = edge + ((size_t)(b * 512 + i) * 512) * 64;
    const float* abase = adj  +  (size_t)(b * 512 + i) * 512;
    float* eout = escore + (size_t)(b * 512 + i) * 512;

    #pragma unroll 1
    for (int jj = 0; jj < 2; ++jj) {
        int j = jj * 256 + tid;
        const float* ep = ebase + (size_t)j * 64;
        __builtin_prefetch(ep + 256 * 64, 0, 1);
        float s = 0.f;
        #pragma unroll
        for (int c = 0; c < 64; c += 4) {
            v4f v = __builtin_nontemporal_load((const v4f*)(ep + c));  // TH=NT: single-use stream
            s += v[0] * wle[c] + v[1] * wle[c + 1] + v[2] * wle[c + 2] + v[3] * wle[c + 3];
        }
        float e = si + s_j[b * 512 + j] + s;
        e = (e > 0.f) ? e : ALPHA_LRELU * e;                 // leaky_relu
        float ad = __builtin_nontemporal_load(abase + j);    // adj used once
        e = (ad > 0.0f) ? e : NEG_INF_F;                     // adjacency mask
        eout[j] = e;
    }
}

// ---------------------------------------------------------------------------
// Kernel 4 (fused): per block = (batch b, 16 i-rows)
//   A: load score tile (32 KB, L2 hit) into padded LDS
//   B: row softmax in LDS
//   C: attention @ h[b] via V_WMMA_F32_16X16X4_F32, ELU, store
// ---------------------------------------------------------------------------
__global__ void attn_kernel(const float* __restrict__ escore, const float* __restrict__ h,
                            float* __restrict__ out) {
    __shared__ float att[16][516];   // padded: stride 516 -> conflict-free column reads
    __shared__ float red[16][16];
    __shared__ float rowstat[16];

    int b   = blockIdx.y;
    int i0  = blockIdx.x * 16;
    int tid = threadIdx.x;

    // ---- Phase A: load scores ------------------------------------------
    const float* ebase = escore + (size_t)(b * 512 + i0) * 512;
    #pragma unroll
    for (int iter = 0; iter < 32; ++iter) {
        int flat = iter * 256 + tid;          // 16*512 (r,j) pairs
        att[flat >> 9][flat & 511] = ebase[flat];
    }
    __syncthreads();

    // ---- Phase B: softmax along j ---------------------------------------
    int row = tid >> 4;
    int l16 = tid & 15;

    float m = -3.0e38f;
    for (int j = l16; j < 512; j += 16) m = fmaxf(m, att[row][j]);
    red[row][l16] = m;
    __syncthreads();
    if (l16 == 0) {
        float mm = red[row][0];
        #pragma unroll
        for (int q = 1; q < 16; ++q) mm = fmaxf(mm, red[row][q]);
        rowstat[row] = mm;
    }
    __syncthreads();
    float mx = rowstat[row];

    float sum = 0.f;
    for (int j = l16; j < 512; j += 16) {
        float v = __expf(att[row][j] - mx);
        att[row][j] = v;
        sum += v;
    }
    red[row][l16] = sum;
    __syncthreads();
    if (l16 == 0) {
        float t = 0.f;
        #pragma unroll
        for (int q = 0; q < 16; ++q) t += red[row][q];
        rowstat[row] = t;
    }
    __syncthreads();
    float inv = 1.0f / rowstat[row];
    for (int j = l16; j < 512; j += 16) att[row][j] *= inv;
    __syncthreads();

    // ---- Phase C: h_prime = attention @ h, then ELU ---------------------
    int wave = tid >> 5;
    int lane = tid & 31;
    int M     = lane & 15;
    int koff  = (lane >> 4) << 1;
    int klo   = (lane < 16) ? 0 : 2;
    int ncol  = lane & 15;
    int mbase = (lane >> 4) << 3;

    const float* hb = h + (size_t)b * 512 * 256;
    float* ob = out + (size_t)(b * 512 + i0) * 256;

    #pragma unroll 1
    for (int t2 = 0; t2 < 2; ++t2) {
        int n0 = (wave + t2 * 8) * 16;
        v8f c = {};
        for (int k0 = 0; k0 < 512; k0 += 4) {
            v2f afrag;                              // ds_load_b64 (8B aligned)
            afrag[0] = att[M][k0 + koff];
            afrag[1] = att[M][k0 + koff + 1];
            v2f bf;
            const float* hp = hb + (size_t)(k0 + klo) * 256 + n0 + ncol;
            bf[0] = hp[0];
            bf[1] = hp[256];
            c = __builtin_amdgcn_wmma_f32_16x16x4_f32(false, afrag, false, bf, (short)0, c, false, false);
        }
        #pragma unroll
        for (int v = 0; v < 8; ++v) {
            float xv = c[v];
            float o = (xv > 0.f) ? xv : (__expf(xv) - 1.0f);   // ELU (alpha=1)
            ob[(size_t)(mbase + v) * 256 + n0 + ncol] = o;
        }
    }
}

// ---------------------------------------------------------------------------
// Inputs (setup_inputs order): 0:x 1:edge 2:adj 3:W 4:W1 5:a   Output: f32
// ---------------------------------------------------------------------------
extern "C" void kernel_launch(void* const* d_in, const int* in_sizes, int n_in,
                              void* d_out, int out_size, void* d_ws, size_t ws_size,
                              hipStream_t stream) {
    const float* x    = (const float*)d_in[0];
    const float* edge = (const float*)d_in[1];
    const float* adj  = (const float*)d_in[2];
    const float* W    = (const float*)d_in[3];
    const float* W1   = (const float*)d_in[4];
    const float* a    = (const float*)d_in[5];
    float* out = (float*)d_out;

    // workspace layout (floats): h | s_i | s_j | w1a3 | escore   (~6.2 MB total)
    float* h      = (float*)d_ws;              // 4*512*256 = 524288
    float* s_i    = h + 4 * 512 * 256;         // 2048
    float* s_j    = s_i + 4 * 512;             // 2048
    float* w1a3   = s_j + 4 * 512;             // 64
    float* escore = w1a3 + 64;                 // 4*512*512 = 1048576

    w1a3_kernel<<<1, 64, 0, stream>>>(W1, a, w1a3);
    gemm_h_kernel<<<dim3(32, 4), 256, 0, stream>>>(x, W, h);
    sij_kernel<<<8, 256, 0, stream>>>(h, a, s_i, s_j);
    score_kernel<<<dim3(512, 4), 256, 0, stream>>>(edge, adj, s_i, s_j, w1a3, escore);
    attn_kernel<<<dim3(32, 4), 256, 0, stream>>>(escore, h, out);
}